// Sae_37134287241674
// MI455X (gfx1250) — compile-verified
//
#include <hip/hip_runtime.h>
#include <hip/hip_bf16.h>

// ---- problem constants -------------------------------------------------
#define BATCH   512
#define NI      2
#define NDIM    512      // in_dim == N
#define DTOT    16384
#define ROWS    (BATCH*NI)                 // 1024
#define TOT     (ROWS*DTOT)                // 16,777,216 acts
#define TOT4    (TOT/4)                    // 4,194,304 float4
#define LVL_STRIDE ((size_t)TOT)           // per-level topk stride (floats)

// workspace layout (u32 units)
#define H1_OFF  0        // 3*2048
#define H2_OFF  6144     // 3*2048
#define H3_OFF  12288    // 3*512
#define SEL_OFF 13824    // [0..2]=bin1 [3..5]=need1 [6..8]=bin2 [9..11]=need2 [12..14]=tauBits

typedef __attribute__((ext_vector_type(2))) float v2f;
typedef __attribute__((ext_vector_type(8))) float v8f;

// ========================================================================
// 1) Encoder GEMM: acts[b,i,h] = relu( sum_d x[b,i,d] * W_enc[i,h,d] + b_enc[i,h] )
//    fp32 WMMA 16x16x4. Block = 256 thr (8 waves) computes 128(M) x 64(N) tile,
//    K staged in LDS in chunks of 32.
// ========================================================================
__global__ __launch_bounds__(256) void sae_encoder(
    const float* __restrict__ x, const float* __restrict__ Wenc,
    const float* __restrict__ benc, float* __restrict__ acts)
{
    __shared__ float Xs[128 * 33];
    __shared__ float Ws[64 * 33];

    const int i    = blockIdx.z;
    const int m0   = blockIdx.y * 128;
    const int n0   = blockIdx.x * 64;
    const int tid  = threadIdx.x;
    const int lane = tid & 31;
    const int wave = tid >> 5;
    const int half = lane >> 4;        // 0 or 1
    const int mr   = lane & 15;

    const float* __restrict__ Wi = Wenc + (size_t)i * DTOT * NDIM;

    v8f acc[4] = {};

    for (int k0 = 0; k0 < NDIM; k0 += 32) {
        // stage X 128x32
        for (int e = tid; e < 128 * 32; e += 256) {
            int r = e >> 5, k = e & 31;
            Xs[r * 33 + k] = x[((size_t)(m0 + r) * NI + i) * NDIM + k0 + k];
        }
        // stage W 64x32 (rows of W_enc are the K dimension: contiguous)
        for (int e = tid; e < 64 * 32; e += 256) {
            int c = e >> 5, k = e & 31;
            Ws[c * 33 + k] = Wi[(size_t)(n0 + c) * NDIM + k0 + k];
        }
        __syncthreads();

        const float* xrow = &Xs[(wave * 16 + mr) * 33];
        #pragma unroll
        for (int kk = 0; kk < 32; kk += 4) {
            v2f a;
            a.x = xrow[kk + 2 * half];
            a.y = xrow[kk + 2 * half + 1];
            #pragma unroll
            for (int ns = 0; ns < 4; ++ns) {
                const float* wr = &Ws[(ns * 16 + mr) * 33];
                v2f b;
                b.x = wr[kk + 2 * half];
                b.y = wr[kk + 2 * half + 1];
                acc[ns] = __builtin_amdgcn_wmma_f32_16x16x4_f32(
                    false, a, false, b, (short)0, acc[ns], false, false);
            }
        }
        __syncthreads();
    }

    // epilogue: C layout VGPR j -> M = j + 8*half, N = mr
    const int mbase = m0 + wave * 16 + half * 8;
    #pragma unroll
    for (int ns = 0; ns < 4; ++ns) {
        const int h = n0 + ns * 16 + mr;
        const float bias = benc[i * DTOT + h];
        #pragma unroll
        for (int j = 0; j < 8; ++j) {
            float v = acc[ns][j] + bias;
            v = v > 0.f ? v : 0.f;
            acts[((size_t)(mbase + j) * NI + i) * DTOT + h] = v;
        }
    }
}

// ========================================================================
// 2) Radix select: 3 histogram passes (bits 30:20, 19:9, 8:0).
//    ReLU output >= 0 -> uint bit order == value order.
// ========================================================================
__global__ __launch_bounds__(256) void k_hist1(const unsigned* __restrict__ acts,
                                               unsigned* __restrict__ hist)
{
    __shared__ unsigned lh[3 * 2048];
    for (int e = threadIdx.x; e < 3 * 2048; e += 256) lh[e] = 0;
    __syncthreads();
    const uint4* a4 = (const uint4*)acts;
    for (unsigned idx = blockIdx.x * 256u + threadIdx.x; idx < TOT4; idx += gridDim.x * 256u) {
        uint4 v = a4[idx];
        unsigned hcol = (idx * 4u) & (DTOT - 1u);
        unsigned t = hcol >> 12; unsigned seg = t > 2u ? 2u : t;     // column segment
        unsigned* bp = &lh[seg * 2048];
        atomicAdd(&bp[(v.x >> 20) & 2047u], 1u);
        atomicAdd(&bp[(v.y >> 20) & 2047u], 1u);
        atomicAdd(&bp[(v.z >> 20) & 2047u], 1u);
        atomicAdd(&bp[(v.w >> 20) & 2047u], 1u);
    }
    __syncthreads();
    for (int e = threadIdx.x; e < 3 * 2048; e += 256) {
        unsigned c = lh[e];
        if (c) atomicAdd(&hist[e], c);
    }
}

__global__ void k_sel1(const unsigned* __restrict__ h, unsigned* __restrict__ sel)
{
    int l = threadIdx.x;
    if (l >= 3) return;
    const unsigned Ks[3] = {65536u, 131072u, 262144u};
    unsigned K = Ks[l], cum = 0, binSel = 0, need = K;
    for (int bin = 2047; bin >= 0; --bin) {
        unsigned c = h[bin];
        if (l >= 1) c += h[2048 + bin];
        if (l >= 2) c += h[4096 + bin];
        if (cum + c >= K) { binSel = (unsigned)bin; need = K - cum; break; }
        cum += c;
    }
    sel[l] = binSel;
    sel[3 + l] = need;
}

__global__ __launch_bounds__(256) void k_hist2(const unsigned* __restrict__ acts,
                                               const unsigned* __restrict__ sel,
                                               unsigned* __restrict__ hist2)
{
    __shared__ unsigned lh[3 * 2048];
    for (int e = threadIdx.x; e < 3 * 2048; e += 256) lh[e] = 0;
    const unsigned b0 = sel[0], b1 = sel[1], b2 = sel[2];
    __syncthreads();
    const uint4* a4 = (const uint4*)acts;
    for (unsigned idx = blockIdx.x * 256u + threadIdx.x; idx < TOT4; idx += gridDim.x * 256u) {
        uint4 v = a4[idx];
        unsigned hcol = (idx * 4u) & (DTOT - 1u);
        unsigned t = hcol >> 12; unsigned lmin = t > 2u ? 2u : t;
        unsigned vs[4] = {v.x, v.y, v.z, v.w};
        #pragma unroll
        for (int q = 0; q < 4; ++q) {
            unsigned top = (vs[q] >> 20) & 2047u;
            unsigned mid = (vs[q] >> 9) & 2047u;
            if (lmin <= 0u && top == b0) atomicAdd(&lh[0 * 2048 + mid], 1u);
            if (lmin <= 1u && top == b1) atomicAdd(&lh[1 * 2048 + mid], 1u);
            if (top == b2)               atomicAdd(&lh[2 * 2048 + mid], 1u);
        }
    }
    __syncthreads();
    for (int e = threadIdx.x; e < 3 * 2048; e += 256) {
        unsigned c = lh[e];
        if (c) atomicAdd(&hist2[e], c);
    }
}

__global__ void k_sel2(const unsigned* __restrict__ h2, unsigned* __restrict__ sel)
{
    int l = threadIdx.x;
    if (l >= 3) return;
    unsigned K = sel[3 + l], cum = 0, b = 0, need = K;
    for (int bin = 2047; bin >= 0; --bin) {
        unsigned c = h2[l * 2048 + bin];
        if (cum + c >= K) { b = (unsigned)bin; need = K - cum; break; }
        cum += c;
    }
    sel[6 + l] = b;
    sel[9 + l] = need;
}

__global__ __launch_bounds__(256) void k_hist3(const unsigned* __restrict__ acts,
                                               const unsigned* __restrict__ sel,
                                               unsigned* __restrict__ hist3)
{
    __shared__ unsigned lh[3 * 512];
    for (int e = threadIdx.x; e < 3 * 512; e += 256) lh[e] = 0;
    const unsigned b0 = sel[0], b1 = sel[1], b2 = sel[2];
    const unsigned m0 = sel[6], m1 = sel[7], m2 = sel[8];
    __syncthreads();
    const uint4* a4 = (const uint4*)acts;
    for (unsigned idx = blockIdx.x * 256u + threadIdx.x; idx < TOT4; idx += gridDim.x * 256u) {
        uint4 v = a4[idx];
        unsigned hcol = (idx * 4u) & (DTOT - 1u);
        unsigned t = hcol >> 12; unsigned lmin = t > 2u ? 2u : t;
        unsigned vs[4] = {v.x, v.y, v.z, v.w};
        #pragma unroll
        for (int q = 0; q < 4; ++q) {
            unsigned top = (vs[q] >> 20) & 2047u;
            unsigned mid = (vs[q] >> 9) & 2047u;
            unsigned low = vs[q] & 511u;
            if (lmin <= 0u && top == b0 && mid == m0) atomicAdd(&lh[0 * 512 + low], 1u);
            if (lmin <= 1u && top == b1 && mid == m1) atomicAdd(&lh[1 * 512 + low], 1u);
            if (top == b2 && mid == m2)               atomicAdd(&lh[2 * 512 + low], 1u);
        }
    }
    __syncthreads();
    for (int e = threadIdx.x; e < 3 * 512; e += 256) {
        unsigned c = lh[e];
        if (c) atomicAdd(&hist3[e], c);
    }
}

__global__ void k_sel3(const unsigned* __restrict__ h3, unsigned* __restrict__ sel)
{
    int l = threadIdx.x;
    if (l >= 3) return;
    unsigned K = sel[9 + l], cum = 0, b = 0;
    for (int bin = 511; bin >= 0; --bin) {
        unsigned c = h3[l * 512 + bin];
        if (cum + c >= K) { b = (unsigned)bin; break; }
        cum += c;
    }
    sel[12 + l] = (sel[l] << 20) | (sel[6 + l] << 9) | b;   // exact K-th largest bit pattern
}

// ========================================================================
// 3) Threshold + write padded topk_levels (3, B, I, D) — pure bandwidth.
// ========================================================================
__global__ __launch_bounds__(256) void k_topk(const float* __restrict__ acts,
                                              const unsigned* __restrict__ sel,
                                              float* __restrict__ topk)
{
    const float tau0 = __uint_as_float(sel[12]);
    const float tau1 = __uint_as_float(sel[13]);
    const float tau2 = __uint_as_float(sel[14]);
    const float4* a4 = (const float4*)acts;
    float4* o4 = (float4*)topk;
    for (unsigned idx = blockIdx.x * 256u + threadIdx.x; idx < TOT4; idx += gridDim.x * 256u) {
        float4 v = a4[idx];
        unsigned hcol = (idx * 4u) & (DTOT - 1u);
        const float tau[3] = {tau0, tau1, tau2};
        #pragma unroll
        for (int l = 0; l < 3; ++l) {
            bool in = hcol < (4096u << l);
            float4 o;
            o.x = (in && v.x >= tau[l]) ? v.x : 0.f;
            o.y = (in && v.y >= tau[l]) ? v.y : 0.f;
            o.z = (in && v.z >= tau[l]) ? v.z : 0.f;
            o.w = (in && v.w >= tau[l]) ? v.w : 0.f;
            o4[(size_t)l * (LVL_STRIDE / 4) + idx] = o;
        }
    }
}

// ========================================================================
// 4) Sparse decode: recon[l,b,i,n] = relu( sum_{h kept} acts[b,i,h]*W_enc[i,h,n] + b_dec[i,n] )
//    (W_dec[i,n,h] == W_enc[i,h,n] -> contiguous rows.)
//    Deterministic wave-private ballot compaction, fixed accumulation order.
// ========================================================================
__global__ __launch_bounds__(256) void k_decode(
    const float* __restrict__ acts, const float* __restrict__ Wenc,
    const float* __restrict__ bdec, const unsigned* __restrict__ sel,
    float* __restrict__ recon)
{
    __shared__ unsigned sh[2048];
    __shared__ float    sv[2048];
    __shared__ unsigned scnt[8];

    const int l    = blockIdx.y;
    const int row  = blockIdx.x;          // b*2 + i
    const int i    = row & 1;
    const int tid  = threadIdx.x;
    const int lane = tid & 31;
    const int wave = tid >> 5;
    const int dl   = 4096 << l;
    const float tau = __uint_as_float(sel[12 + l]);

    const float* __restrict__ arow = acts + (size_t)row * DTOT;
    const float* __restrict__ Wi   = Wenc + (size_t)i * DTOT * NDIM;

    float acc0 = 0.f, acc1 = 0.f;

    for (int c0 = 0; c0 < dl; c0 += 2048) {
        // each wave compacts its own 256-column slab -> deterministic order
        unsigned base = 0;
        const int colBase = c0 + wave * 256;
        #pragma unroll
        for (int j = 0; j < 8; ++j) {
            int h = colBase + j * 32 + lane;
            float v = arow[h];
            bool p = (v >= tau) && (v > 0.f);
            unsigned mask = __builtin_amdgcn_ballot_w32(p);
            if (p) {
                unsigned pos = __popc(mask & ((1u << lane) - 1u));
                sh[wave * 256 + base + pos] = (unsigned)h;
                sv[wave * 256 + base + pos] = v;
            }
            base += (unsigned)__popc(mask);
        }
        if (lane == 0) scnt[wave] = base;
        __syncthreads();

        #pragma unroll 1
        for (int w = 0; w < 8; ++w) {
            const unsigned n = scnt[w];
            for (unsigned j = 0; j < n; ++j) {
                const unsigned h = sh[w * 256 + j];
                const float    v = sv[w * 256 + j];
                const float* wr = Wi + (size_t)h * NDIM;   // contiguous 2KB row
                acc0 = fmaf(v, wr[tid], acc0);
                acc1 = fmaf(v, wr[tid + 256], acc1);
            }
        }
        __syncthreads();
    }

    float r0 = acc0 + bdec[i * NDIM + tid];
    float r1 = acc1 + bdec[i * NDIM + tid + 256];
    r0 = r0 > 0.f ? r0 : 0.f;
    r1 = r1 > 0.f ? r1 : 0.f;
    const size_t obase = ((size_t)l * ROWS + row) * NDIM;
    recon[obase + tid] = r0;
    recon[obase + tid + 256] = r1;
}

// ========================================================================
extern "C" void kernel_launch(void* const* d_in, const int* in_sizes, int n_in,
                              void* d_out, int out_size, void* d_ws, size_t ws_size,
                              hipStream_t stream)
{
    const float* x    = (const float*)d_in[0];
    const float* Wenc = (const float*)d_in[1];
    // d_in[2] = W_dec is W_enc transposed; we read W_enc rows instead (contiguous)
    const float* benc = (const float*)d_in[3];
    const float* bdec = (const float*)d_in[4];

    float* out   = (float*)d_out;
    float* recon = out;                                      // (3,B,I,N)
    float* topk  = out + (size_t)3 * ROWS * NDIM;            // (3,B,I,D)
    float* acts  = topk + (size_t)3 * LVL_STRIDE;            // (B,I,D)

    unsigned* ws = (unsigned*)d_ws;

    hipMemsetAsync(d_ws, 0, 64 * 1024, stream);

    sae_encoder<<<dim3(DTOT / 64, BATCH / 128, NI), 256, 0, stream>>>(x, Wenc, benc, acts);

    k_hist1<<<1024, 256, 0, stream>>>((const unsigned*)acts, ws + H1_OFF);
    k_sel1 <<<1, 32, 0, stream>>>(ws + H1_OFF, ws + SEL_OFF);
    k_hist2<<<1024, 256, 0, stream>>>((const unsigned*)acts, ws + SEL_OFF, ws + H2_OFF);
    k_sel2 <<<1, 32, 0, stream>>>(ws + H2_OFF, ws + SEL_OFF);
    k_hist3<<<1024, 256, 0, stream>>>((const unsigned*)acts, ws + SEL_OFF, ws + H3_OFF);
    k_sel3 <<<1, 32, 0, stream>>>(ws + H3_OFF, ws + SEL_OFF);

    k_topk  <<<2048, 256, 0, stream>>>(acts, ws + SEL_OFF, topk);
    k_decode<<<dim3(ROWS, 3), 256, 0, stream>>>(acts, Wenc, bdec, ws + SEL_OFF, recon);
}